// TGN_67104569033096
// MI455X (gfx1250) — compile-verified
//
#include <hip/hip_runtime.h>
#include <math.h>

// ---------------------------------------------------------------------------
// TGN heterogeneous-attention embedding for MI455X (gfx1250), fp32 WMMA path.
// All 32x32 GEMMs run on V_WMMA_F32_16X16X4_F32 (16-edge/node tiles per wave).
// Weights are staged TRANSPOSED in LDS so each B operand is a contiguous,
// 8-byte-aligned pair -> single ds_load_b64 per operand (no marshalling movs).
// ---------------------------------------------------------------------------

#define WPB 8            // waves per block
#define BT  256          // block threads (wave32 * 8)

typedef float v2f __attribute__((ext_vector_type(2)));
typedef float v8f __attribute__((ext_vector_type(8)));

// order-preserving f32 -> u32 mapping for atomicMax-based segment max
__device__ __forceinline__ unsigned fenc(float f) {
  unsigned u = __float_as_uint(f);
  return (u & 0x80000000u) ? ~u : (u | 0x80000000u);
}
__device__ __forceinline__ float fdec(unsigned k) {
  unsigned u = (k & 0x80000000u) ? (k ^ 0x80000000u) : ~k;
  return __uint_as_float(u);
}

// stage a 32x32 row-major global weight matrix into LDS *transposed*
__device__ __forceinline__ void stage_weightT(const float* __restrict__ W, float* sWt) {
  for (int i = threadIdx.x; i < 1024; i += BT) {
    const int r = i >> 5, c = i & 31;
    sWt[c * 32 + r] = W[i];
  }
}

// [16x32](LDS,row-major) @ [32x32] (LDS, TRANSPOSED: Wt[col][row]) -> two 16x16 tiles.
// A layout (16x4 f32): lane m=l&15; VGPR0 = K(4kk+2hi), VGPR1 = K(4kk+2hi+1).
// B layout (4x16 f32): lane n=l&15; row = 4kk + 2hi (+1 for 2nd VGPR), col = 16t+n.
__device__ __forceinline__ void wmma_16x32x32(const float* A, const float* Wt, v8f acc[2]) {
  const int lane = threadIdx.x & 31;
  const int m  = lane & 15;
  const int hi = lane >> 4;
  v8f z = {0.f, 0.f, 0.f, 0.f, 0.f, 0.f, 0.f, 0.f};
  acc[0] = z; acc[1] = z;
#pragma unroll
  for (int kk = 0; kk < 8; ++kk) {
    const int k0 = kk * 4 + hi * 2;                       // even -> 8B aligned
    v2f a  = *(const v2f*)&A[m * 32 + k0];                // A[m][k0], A[m][k0+1]
    v2f b0 = *(const v2f*)&Wt[m * 32 + k0];               // W[k0][m],    W[k0+1][m]
    v2f b1 = *(const v2f*)&Wt[(m + 16) * 32 + k0];        // W[k0][m+16], W[k0+1][m+16]
    acc[0] = __builtin_amdgcn_wmma_f32_16x16x4_f32(false, a, false, b0, (short)0, acc[0], false, false);
    acc[1] = __builtin_amdgcn_wmma_f32_16x16x4_f32(false, a, false, b1, (short)0, acc[1], false, false);
  }
}

// D layout: lane n=l&15, hi=l>>4; VGPR r holds row (8*hi + r), col 16t+n.
__device__ __forceinline__ void wmma_store_lds(const v8f acc[2], float* Out) {
  const int lane = threadIdx.x & 31;
  const int n  = lane & 15;
  const int hi = lane >> 4;
#pragma unroll
  for (int r = 0; r < 8; ++r) {
    Out[(hi * 8 + r) * 32 + n]      = acc[0][r];
    Out[(hi * 8 + r) * 32 + 16 + n] = acc[1][r];
  }
}

// coalesced gather of 16 rows (32 f32 each) into LDS tile
__device__ __forceinline__ void gather_rows16(const float* __restrict__ base,
                                              const int* rows, float* dst) {
  const int lane = threadIdx.x & 31;
#pragma unroll
  for (int i = 0; i < 16; ++i)
    dst[i * 32 + lane] = base[(size_t)rows[i] * 32 + lane];
}

// ---------------------------------------------------------------------------
__global__ void fill_f32_kernel(float* __restrict__ p, float v, int n) {
  int t = blockIdx.x * blockDim.x + threadIdx.x;
  if (t < n) p[t] = v;
}
__global__ void fill_u32_kernel(unsigned* __restrict__ p, unsigned v, int n) {
  int t = blockIdx.x * blockDim.x + threadIdx.x;
  if (t < n) p[t] = v;
}

// h0 = memory[gather] + cos(ts*w + b); also emit gathered memory to output
__global__ void init_nodes_kernel(const float* __restrict__ mem, const int* __restrict__ nx,
                                  const float* __restrict__ ts, const float* __restrict__ tw,
                                  const float* __restrict__ tb, float* __restrict__ h0,
                                  float* __restrict__ mem_out, int nb) {
  int t = blockIdx.x * blockDim.x + threadIdx.x;
  if (t >= nb * 32) return;
  int i = t >> 5, c = t & 31;
  float m = mem[(size_t)nx[i] * 32 + c];
  mem_out[t] = m;
  h0[t] = m + __cosf(ts[i] * tw[c] + tb[c]);
}

// Pass A: s[e,h] = (q.k)/4 (masked -> -1e9); atomicMax segment max per dst.
__global__ void __launch_bounds__(BT)
attn_score_kernel(const float* __restrict__ hsrc, const float* __restrict__ hdst,
                  const int* __restrict__ src, const int* __restrict__ dst,
                  const int* __restrict__ mask,
                  const float* __restrict__ Wk, const float* __restrict__ Wq,
                  float* __restrict__ s_out, unsigned* __restrict__ smax, int nE) {
  __shared__ __align__(16) float sWk[1024], sWq[1024];
  __shared__ __align__(16) float sS[WPB][512];   // src rows, then reused for q*k product
  __shared__ __align__(16) float sD[WPB][512];   // dst rows
  __shared__ int sIdx[WPB][32];                  // [0..15]=src, [16..31]=dst

  const int tid = threadIdx.x;
  stage_weightT(Wk, sWk);
  stage_weightT(Wq, sWq);

  const int w = tid >> 5, lane = tid & 31;
  const int ntiles = (nE + 15) >> 4;
  const int wid = blockIdx.x * WPB + w;
  const bool valid = wid < ntiles;
  const int tile = valid ? wid : 0;
  const int e0 = tile * 16;

  if (lane < 16) sIdx[w][lane] = src[min(e0 + lane, nE - 1)];
  else           sIdx[w][lane] = dst[min(e0 + lane - 16, nE - 1)];
  __syncthreads();

  gather_rows16(hsrc, &sIdx[w][0],  sS[w]);
  gather_rows16(hdst, &sIdx[w][16], sD[w]);
  __syncthreads();

  v8f k[2], q[2];
  wmma_16x32x32(sS[w], sWk, k);
  wmma_16x32x32(sD[w], sWq, q);
  __syncthreads();

  { // stage elementwise q*k back into sS (D layout -> row-major tile)
    const int n = lane & 15, hi = lane >> 4;
#pragma unroll
    for (int r = 0; r < 8; ++r) {
      sS[w][(hi * 8 + r) * 32 + n]      = k[0][r] * q[0][r];
      sS[w][(hi * 8 + r) * 32 + 16 + n] = k[1][r] * q[1][r];
    }
  }
  __syncthreads();

  // 32 lanes -> 16 edges x 2 heads
  const int el = lane >> 1, h = lane & 1;
  float s = 0.f;
#pragma unroll
  for (int j = 0; j < 16; ++j) s += sS[w][el * 32 + h * 16 + j];
  s *= 0.25f;                                   // / sqrt(DH=16)
  const int e = e0 + el;
  if (valid && e < nE) {
    if (mask[e] == 0) s = -1e9f;
    s_out[(size_t)e * 2 + h] = s;
    atomicMax(&smax[(size_t)sIdx[w][16 + el] * 2 + h], fenc(s));
  }
}

// Pass B: den[dst,h] += exp(s - smax[dst,h]) for unmasked edges (den init 1e-9)
__global__ void denom_kernel(const float* __restrict__ s_buf, const int* __restrict__ mask,
                             const int* __restrict__ dst, const unsigned* __restrict__ smax,
                             float* __restrict__ den, int nE) {
  int t = blockIdx.x * blockDim.x + threadIdx.x;
  if (t >= nE * 2) return;
  int e = t >> 1, h = t & 1;
  if (mask[e] == 0) return;
  int d = dst[e];
  float p = __expf(s_buf[t] - fdec(smax[(size_t)d * 2 + h]));
  atomicAdd(&den[(size_t)d * 2 + h], p);
}

// Pass C: v = (hs + ef@elin + elb + cos-time-enc) @ Wv ; agg[dst] += w * v
__global__ void __launch_bounds__(BT)
aggregate_kernel(const float* __restrict__ hsrc, const float* __restrict__ efeat,
                 const int* __restrict__ eidx, const float* __restrict__ relt,
                 const int* __restrict__ src, const int* __restrict__ dst,
                 const int* __restrict__ mask,
                 const float* __restrict__ elw, const float* __restrict__ elb,
                 const float* __restrict__ Wv,
                 const float* __restrict__ tw, const float* __restrict__ tb,
                 const float* __restrict__ s_buf, const unsigned* __restrict__ smax,
                 const float* __restrict__ den,
                 float* __restrict__ agg, int nE) {
  __shared__ __align__(16) float sWe[1024], sWv[1024];
  __shared__ float sTw[32], sTb[32], sEb[32];
  __shared__ __align__(16) float bufX[WPB][512];
  __shared__ __align__(16) float bufH[WPB][512];
  __shared__ float sRel[WPB][16];
  __shared__ float sWt[WPB][32];
  __shared__ int   sIdx[WPB][48];   // [0..15]=eidx rows, [16..31]=src, [32..47]=dst

  const int tid = threadIdx.x;
  stage_weightT(elw, sWe);
  stage_weightT(Wv, sWv);
  if (tid < 32) { sTw[tid] = tw[tid]; sTb[tid] = tb[tid]; sEb[tid] = elb[tid]; }

  const int w = tid >> 5, lane = tid & 31;
  const int ntiles = (nE + 15) >> 4;
  const int wid = blockIdx.x * WPB + w;
  const bool valid = wid < ntiles;
  const int tile = valid ? wid : 0;
  const int e0 = tile * 16;

  if (lane < 16) {
    int e = min(e0 + lane, nE - 1);
    sIdx[w][lane]      = eidx[e];
    sIdx[w][32 + lane] = dst[e];
    sRel[w][lane]      = relt[e];
  } else {
    int e = min(e0 + lane - 16, nE - 1);
    sIdx[w][lane] = src[e];
  }
  __syncthreads();

  gather_rows16(efeat, &sIdx[w][0],  bufX[w]);   // edge features (F=32)
  gather_rows16(hsrc,  &sIdx[w][16], bufH[w]);   // h_src rows
  __syncthreads();

  v8f accE[2];
  wmma_16x32x32(bufX[w], sWe, accE);             // ef @ elin
  __syncthreads();

  { // bufX = ef@elin + elb + cos(rel_t*w+b) + hs  (stage in D layout)
    const int n = lane & 15, hi = lane >> 4;
#pragma unroll
    for (int r = 0; r < 8; ++r) {
      const int m = hi * 8 + r;
#pragma unroll
      for (int t2 = 0; t2 < 2; ++t2) {
        const int c = t2 * 16 + n;
        const float te = __cosf(sRel[w][m] * sTw[c] + sTb[c]);
        bufX[w][m * 32 + c] = accE[t2][r] + sEb[c] + te + bufH[w][m * 32 + c];
      }
    }
  }
  __syncthreads();

  v8f accV[2];
  wmma_16x32x32(bufX[w], sWv, accV);             // (hs+ef+te) @ Wv
  __syncthreads();
  wmma_store_lds(accV, bufH[w]);                 // v tile -> bufH

  { // softmax weight per (edge, head)
    const int el = lane >> 1, h = lane & 1;
    const int e = e0 + el;
    float wt = 0.f;
    if (e < nE && mask[e] != 0) {
      const int d = sIdx[w][32 + el];
      const float s = s_buf[(size_t)e * 2 + h];
      wt = __expf(s - fdec(smax[(size_t)d * 2 + h])) / den[(size_t)d * 2 + h];
    }
    sWt[w][el * 2 + h] = wt;
  }
  __syncthreads();

  if (valid) {
#pragma unroll 4
    for (int r = 0; r < 16; ++r) {
      const int e = e0 + r;
      if (e >= nE) break;
      const float wt = sWt[w][r * 2 + (lane >> 4)];
      if (wt != 0.f) {
        const int d = sIdx[w][32 + r];
        atomicAdd(&agg[(size_t)d * 32 + lane], wt * bufH[w][r * 32 + lane]);
      }
    }
  }
}

// h_new = relu(agg @ lin) + h_old
__global__ void __launch_bounds__(BT)
update_kernel(const float* __restrict__ agg, const float* __restrict__ lin,
              const float* __restrict__ h_old, float* __restrict__ h_new, int nb) {
  __shared__ __align__(16) float sL[1024];
  __shared__ __align__(16) float bufA[WPB][512];
  const int tid = threadIdx.x;
  stage_weightT(lin, sL);

  const int w = tid >> 5, lane = tid & 31;
  const int ntiles = (nb + 15) >> 4;
  const int wid = blockIdx.x * WPB + w;
  const bool valid = wid < ntiles;
  const int tile = valid ? wid : 0;
  const int n0 = tile * 16;

#pragma unroll
  for (int i = 0; i < 16; ++i) {
    const int rr = min(n0 + i, nb - 1);
    bufA[w][i * 32 + lane] = agg[(size_t)rr * 32 + lane];
  }
  __syncthreads();

  v8f acc[2];
  wmma_16x32x32(bufA[w], sL, acc);

  if (valid) {
    const int n = lane & 15, hi = lane >> 4;
#pragma unroll
    for (int r = 0; r < 8; ++r) {
      const int row = n0 + hi * 8 + r;
      if (row < nb) {
        const size_t i0 = (size_t)row * 32 + n;
        h_new[i0]      = fmaxf(acc[0][r], 0.f) + h_old[i0];
        h_new[i0 + 16] = fmaxf(acc[1][r], 0.f) + h_old[i0 + 16];
      }
    }
  }
}

// ---------------------------------------------------------------------------
extern "C" void kernel_launch(void* const* d_in, const int* in_sizes, int n_in,
                              void* d_out, int out_size, void* d_ws, size_t ws_size,
                              hipStream_t stream) {
  (void)n_in; (void)out_size; (void)ws_size;

  const float* memA  = (const float*)d_in[0];
  const float* memP  = (const float*)d_in[1];
  const float* efeat = (const float*)d_in[2];
  const int*   nxA   = (const int*)d_in[3];
  const int*   nxP   = (const int*)d_in[4];
  const float* tsA   = (const float*)d_in[5];
  const float* tsP   = (const float*)d_in[6];
  const int*   eiW   = (const int*)d_in[7];
  const float* rtW   = (const float*)d_in[8];
  const int*   exW   = (const int*)d_in[9];
  const int*   mkW   = (const int*)d_in[10];
  const int*   eiR   = (const int*)d_in[11];
  const float* rtR   = (const float*)d_in[12];
  const int*   exR   = (const int*)d_in[13];
  const int*   mkR   = (const int*)d_in[14];
  const float* tw    = (const float*)d_in[15];
  const float* tb    = (const float*)d_in[16];
  const float* WkW   = (const float*)d_in[17];
  const float* WqW   = (const float*)d_in[18];
  const float* WvW   = (const float*)d_in[19];
  const float* elwW  = (const float*)d_in[20];
  const float* elbW  = (const float*)d_in[21];
  const float* WkR   = (const float*)d_in[22];
  const float* WqR   = (const float*)d_in[23];
  const float* WvR   = (const float*)d_in[24];
  const float* elwR  = (const float*)d_in[25];
  const float* elbR  = (const float*)d_in[26];
  const float* linA  = (const float*)d_in[27];
  const float* linP  = (const float*)d_in[28];

  const int NBn = in_sizes[3];       // nodes per type
  const int En  = in_sizes[8];       // edges per type

  // workspace carve (~82 MB): h ping-pong, edge scores, segmax/den, one agg buf
  char* p = (char*)d_ws;
  auto carve = [&](size_t bytes) -> void* {
    void* q = (void*)p; p += (bytes + 255) & ~(size_t)255; return q;
  };
  float*    hA0  = (float*)carve((size_t)NBn * 32 * 4);
  float*    hP0  = (float*)carve((size_t)NBn * 32 * 4);
  float*    hA1  = (float*)carve((size_t)NBn * 32 * 4);
  float*    hP1  = (float*)carve((size_t)NBn * 32 * 4);
  float*    sbuf = (float*)carve((size_t)En * 2 * 4);
  unsigned* smax = (unsigned*)carve((size_t)NBn * 2 * 4);
  float*    den  = (float*)carve((size_t)NBn * 2 * 4);
  float*    agg  = (float*)carve((size_t)NBn * 32 * 4);

  float* out     = (float*)d_out;
  float* hAout   = out;
  float* hPout   = out + (size_t)NBn * 32;
  float* memAout = out + 2 * (size_t)NBn * 32;
  float* memPout = out + 3 * (size_t)NBn * 32;

  const int nElem  = NBn * 32;
  const int etiles = (En + 15) / 16;
  const int eblk   = (etiles + WPB - 1) / WPB;
  const int ntls   = (NBn + 15) / 16;
  const int nblk   = (ntls + WPB - 1) / WPB;

  init_nodes_kernel<<<(nElem + 255) / 256, 256, 0, stream>>>(memA, nxA, tsA, tw, tb, hA0, memAout, NBn);
  init_nodes_kernel<<<(nElem + 255) / 256, 256, 0, stream>>>(memP, nxP, tsP, tw, tb, hP0, memPout, NBn);

  for (int layer = 0; layer < 2; ++layer) {
    const float* hAc = layer ? hA1 : hA0;
    const float* hPc = layer ? hP1 : hP0;
    float* hAn = layer ? hAout : hA1;
    float* hPn = layer ? hPout : hP1;

    // ---- edge type "writes": author(src) -> paper(dst), updates h_p
    fill_u32_kernel<<<(NBn * 2 + 255) / 256, 256, 0, stream>>>(smax, 0u, NBn * 2);
    fill_f32_kernel<<<(NBn * 2 + 255) / 256, 256, 0, stream>>>(den, 1e-9f, NBn * 2);
    fill_f32_kernel<<<(nElem + 255) / 256, 256, 0, stream>>>(agg, 0.f, nElem);
    attn_score_kernel<<<eblk, BT, 0, stream>>>(hAc, hPc, eiW, eiW + En, mkW, WkW, WqW, sbuf, smax, En);
    denom_kernel<<<(2 * En + 255) / 256, 256, 0, stream>>>(sbuf, mkW, eiW + En, smax, den, En);
    aggregate_kernel<<<eblk, BT, 0, stream>>>(hAc, efeat, exW, rtW, eiW, eiW + En, mkW,
                                              elwW, elbW, WvW, tw, tb, sbuf, smax, den, agg, En);
    update_kernel<<<nblk, BT, 0, stream>>>(agg, linP, hPc, hPn, NBn);

    // ---- edge type "rev": paper(src) -> author(dst), updates h_a
    fill_u32_kernel<<<(NBn * 2 + 255) / 256, 256, 0, stream>>>(smax, 0u, NBn * 2);
    fill_f32_kernel<<<(NBn * 2 + 255) / 256, 256, 0, stream>>>(den, 1e-9f, NBn * 2);
    fill_f32_kernel<<<(nElem + 255) / 256, 256, 0, stream>>>(agg, 0.f, nElem);
    attn_score_kernel<<<eblk, BT, 0, stream>>>(hPc, hAc, eiR, eiR + En, mkR, WkR, WqR, sbuf, smax, En);
    denom_kernel<<<(2 * En + 255) / 256, 256, 0, stream>>>(sbuf, mkR, eiR + En, smax, den, En);
    aggregate_kernel<<<eblk, BT, 0, stream>>>(hPc, efeat, exR, rtR, eiR, eiR + En, mkR,
                                              elwR, elbR, WvR, tw, tb, sbuf, smax, den, agg, En);
    update_kernel<<<nblk, BT, 0, stream>>>(agg, linA, hAc, hAn, NBn);
  }
}